// osc_loss_19370302505302
// MI455X (gfx1250) — compile-verified
//
#include <hip/hip_runtime.h>

typedef __attribute__((ext_vector_type(4))) float v4f;
typedef __attribute__((ext_vector_type(2))) float v2f;
typedef __attribute__((ext_vector_type(8))) float v8f;

#define NBLK 2048   // multiple of 64 (required by stage-2 WMMA A-layout mapping)
#define TPB  256    // 8 wave32s per block

// Stage 1: bandwidth-bound streaming pass.
// Each aligned float4 covers global indices (4i, 4i+1, 4i+2, 4i+3) = (even,odd,even,odd),
// so the contribution is max(-x,0), max(+y,0), max(-z,0), max(+w,0) — negates fold into
// v_max_f32 source modifiers. 4 independent b128 NT loads in flight per lane.
__global__ __launch_bounds__(TPB) void osc_loss_stage1(const float* __restrict__ res,
                                                       float* __restrict__ partial,
                                                       int n4) {
    const v4f* __restrict__ p = (const v4f*)res;
    const int stride = NBLK * TPB;
    int i = blockIdx.x * TPB + threadIdx.x;

    float acc0 = 0.f, acc1 = 0.f, acc2 = 0.f, acc3 = 0.f;

    // main loop: 4 outstanding 128-bit non-temporal loads per iteration
    for (; i + 3 * stride < n4; i += 4 * stride) {
        v4f v0 = __builtin_nontemporal_load(p + i);
        v4f v1 = __builtin_nontemporal_load(p + i + stride);
        v4f v2 = __builtin_nontemporal_load(p + i + 2 * stride);
        v4f v3 = __builtin_nontemporal_load(p + i + 3 * stride);
        acc0 += fmaxf(-v0.x, 0.f) + fmaxf(v0.y, 0.f) + fmaxf(-v0.z, 0.f) + fmaxf(v0.w, 0.f);
        acc1 += fmaxf(-v1.x, 0.f) + fmaxf(v1.y, 0.f) + fmaxf(-v1.z, 0.f) + fmaxf(v1.w, 0.f);
        acc2 += fmaxf(-v2.x, 0.f) + fmaxf(v2.y, 0.f) + fmaxf(-v2.z, 0.f) + fmaxf(v2.w, 0.f);
        acc3 += fmaxf(-v3.x, 0.f) + fmaxf(v3.y, 0.f) + fmaxf(-v3.z, 0.f) + fmaxf(v3.w, 0.f);
    }
    // remainder float4s
    for (; i < n4; i += stride) {
        v4f v = __builtin_nontemporal_load(p + i);
        acc0 += fmaxf(-v.x, 0.f) + fmaxf(v.y, 0.f) + fmaxf(-v.z, 0.f) + fmaxf(v.w, 0.f);
    }

    float acc = (acc0 + acc1) + (acc2 + acc3);

    // wave32 reduction
    for (int off = 16; off > 0; off >>= 1)
        acc += __shfl_xor(acc, off, 32);

    __shared__ float wsum[TPB / 32];
    const int lane = threadIdx.x & 31;
    const int wid  = threadIdx.x >> 5;
    if (lane == 0) wsum[wid] = acc;
    __syncthreads();

    if (threadIdx.x < TPB / 32) {
        float v = wsum[threadIdx.x];
        for (int off = 4; off > 0; off >>= 1)   // 8 wave sums live in lanes 0..7
            v += __shfl_xor(v, off, 32);
        if (threadIdx.x == 0) partial[blockIdx.x] = v;
    }
}

// Stage 2: one wave32 reduces NBLK partials deterministically via
// V_WMMA_F32_16X16X4_F32 with B = ones, C = 0.
// A (16x4 f32) documented layout: lane L<16 -> M=L, VGPR0:K=0, VGPR1:K=1;
//                                 lane L>=16 -> M=L-16, VGPR0:K=2, VGPR1:K=3.
// Pre-accumulate partial[64j + m*4 + k] into slot (m,k). Then
// D[m][n] = rowsum(A)[m]; per the documented C/D layout lane 0's 8 VGPRs hold
// rowsums 0..7 and lane 16's hold rowsums 8..15. Sum 8 components + one
// shfl_xor(16) yields the total on every lane.
__global__ __launch_bounds__(32) void osc_loss_stage2(const float* __restrict__ partial,
                                                      const float* __restrict__ res,
                                                      float* __restrict__ out,
                                                      int nblk, int n4, int n) {
    const int lane = threadIdx.x;            // 0..31, EXEC all ones for WMMA
    const int m  = lane & 15;
    const int k0 = (lane < 16) ? 0 : 2;

    float a0 = 0.f, a1 = 0.f;
    for (int j = 0; j < nblk; j += 64) {
        const int base = j + m * 4 + k0;
        a0 += partial[base];
        a1 += partial[base + 1];
    }

    // fold in any scalar tail (n not divisible by 4) in lane 0's (m=0,k=0) slot
    float tail = 0.f;
    for (int t = n4 * 4; t < n; ++t) {
        float x = res[t];
        tail += ((t & 1) == 0) ? fmaxf(-x, 0.f) : fmaxf(x, 0.f);
    }
    if (lane == 0) a0 += tail;

    v2f A; A.x = a0;  A.y = a1;
    v2f B; B.x = 1.f; B.y = 1.f;
    v8f C = {};
    // D = A x ones + 0  ->  D[m][n] = rowsum(A)[m]
    C = __builtin_amdgcn_wmma_f32_16x16x4_f32(
        /*neg_a=*/false, A, /*neg_b=*/false, B,
        /*c_mod=*/(short)0, C, /*reuse_a=*/false, /*reuse_b=*/false);

    float s = ((C[0] + C[1]) + (C[2] + C[3])) + ((C[4] + C[5]) + (C[6] + C[7]));
    s += __shfl_xor(s, 16, 32);              // rowsums 0..7 (+) rowsums 8..15

    if (lane == 0) out[0] = s;
}

extern "C" void kernel_launch(void* const* d_in, const int* in_sizes, int n_in,
                              void* d_out, int out_size, void* d_ws, size_t ws_size,
                              hipStream_t stream) {
    const float* res = (const float*)d_in[0];
    float* out       = (float*)d_out;
    float* partial   = (float*)d_ws;         // NBLK floats = 8 KB scratch

    const int n  = in_sizes[0];
    const int n4 = n >> 2;

    osc_loss_stage1<<<NBLK, TPB, 0, stream>>>(res, partial, n4);
    osc_loss_stage2<<<1, 32, 0, stream>>>(partial, res, out, NBLK, n4, n);
}